// GCNEncoder_65051574665788
// MI455X (gfx1250) — compile-verified
//
#include <hip/hip_runtime.h>
#include <hip/hip_bf16.h>

#define N_NODES 50000
#define N_EDGES 600000
#define N_GRAPHS 128
#define EMB 128
#define LAYERS 5
#define BN_EPS 1e-5f

typedef __attribute__((ext_vector_type(2))) float v2f;
typedef __attribute__((ext_vector_type(8))) float v8f;

// ---------------------------------------------------------------- utilities
__global__ void zero_f32(float* __restrict__ p, int n) {
    int i = blockIdx.x * blockDim.x + threadIdx.x;
    if (i < n) p[i] = 0.0f;
}

__global__ void deg_kernel(const int* __restrict__ dst, float* __restrict__ deg, int E) {
    int i = blockIdx.x * blockDim.x + threadIdx.x;
    if (i < E) atomicAdd(&deg[dst[i]], 1.0f);
}

__global__ void invdenom_kernel(float* __restrict__ deg, int n) {
    int i = blockIdx.x * blockDim.x + threadIdx.x;
    if (i < n) deg[i] = 1.0f / fmaxf(deg[i], 1.0f);
}

// ------------------------------------------------- node encoder: h = x@W + b
// M=50000 (divisible by 16), K=48, N=128. One wave per 16x16 tile,
// 8 waves/block cover all 8 N-tiles.  fp32 WMMA keeps reference numerics.
__global__ void __launch_bounds__(256)
node_encoder_gemm(const float* __restrict__ X,   // [M,48]
                  const float* __restrict__ W,   // [48,128]
                  const float* __restrict__ B,   // [128]
                  float* __restrict__ H)         // [M,128]
{
    const int lane = threadIdx.x & 31;
    const int wave = threadIdx.x >> 5;
    const int half = lane >> 4;          // 0: K pair {0,1}, 1: K pair {2,3}
    const int am   = lane & 15;          // A-row / B-col within tile
    const int m0   = blockIdx.x * 16;
    const int n0   = wave * 16;
    const int row  = m0 + am;
    const int col  = n0 + am;

    v8f acc;
    const float bias = B[col];
#pragma unroll
    for (int r = 0; r < 8; ++r) acc[r] = bias;

#pragma unroll
    for (int k0 = 0; k0 < 48; k0 += 4) {
        const int ka = k0 + 2 * half;
        v2f a, b;
        a.x = X[row * 48 + ka];
        a.y = X[row * 48 + ka + 1];
        b.x = W[ka * EMB + col];
        b.y = W[(ka + 1) * EMB + col];
        acc = __builtin_amdgcn_wmma_f32_16x16x4_f32(
            false, a, false, b, (short)0, acc, false, false);
    }

#pragma unroll
    for (int r = 0; r < 8; ++r)
        H[(m0 + r + 8 * half) * EMB + col] = acc[r];
}

// ------------- fused layer GEMM: Hout = (AGG * invd_row)@Wl + H@Wr + bl
__global__ void __launch_bounds__(256)
layer_gemm(const float* __restrict__ AGG,   // [M,128]
           const float* __restrict__ INVD,  // [M]
           const float* __restrict__ H,     // [M,128]
           const float* __restrict__ Wl,    // [128,128]
           const float* __restrict__ Wr,    // [128,128]
           const float* __restrict__ BL,    // [128]
           float* __restrict__ Hout)        // [M,128]
{
    const int lane = threadIdx.x & 31;
    const int wave = threadIdx.x >> 5;
    const int half = lane >> 4;
    const int am   = lane & 15;
    const int m0   = blockIdx.x * 16;
    const int n0   = wave * 16;
    const int row  = m0 + am;
    const int col  = n0 + am;

    v8f acc;
    const float bias = BL[col];
#pragma unroll
    for (int r = 0; r < 8; ++r) acc[r] = bias;

    const float invd = INVD[row];   // fold mean denominator into A-load

#pragma unroll 4
    for (int k0 = 0; k0 < EMB; k0 += 4) {          // agg @ Wl
        const int ka = k0 + 2 * half;
        v2f a, b;
        a.x = AGG[row * EMB + ka] * invd;
        a.y = AGG[row * EMB + ka + 1] * invd;
        b.x = Wl[ka * EMB + col];
        b.y = Wl[(ka + 1) * EMB + col];
        acc = __builtin_amdgcn_wmma_f32_16x16x4_f32(
            false, a, false, b, (short)0, acc, false, false);
    }
#pragma unroll 4
    for (int k0 = 0; k0 < EMB; k0 += 4) {          // h @ Wr
        const int ka = k0 + 2 * half;
        v2f a, b;
        a.x = H[row * EMB + ka];
        a.y = H[row * EMB + ka + 1];
        b.x = Wr[ka * EMB + col];
        b.y = Wr[(ka + 1) * EMB + col];
        acc = __builtin_amdgcn_wmma_f32_16x16x4_f32(
            false, a, false, b, (short)0, acc, false, false);
    }

#pragma unroll
    for (int r = 0; r < 8; ++r)
        Hout[(m0 + r + 8 * half) * EMB + col] = acc[r];
}

// --------- per-edge: recompute edge encoder, gather h[src], scatter to agg
// One wave per edge; each lane owns 4 channels.
__global__ void __launch_bounds__(256)
scatter_msgs(const int* __restrict__ src, const int* __restrict__ dst,
             const float* __restrict__ eattr,   // [E,11]
             const float* __restrict__ ew,      // [E]
             const float* __restrict__ Wb,      // [11,128]
             const float* __restrict__ bb,      // [128]
             const float* __restrict__ H,       // [N,128]
             float* __restrict__ AGG, int E)
{
    const int e    = (blockIdx.x * blockDim.x + threadIdx.x) >> 5;
    const int lane = threadIdx.x & 31;
    if (e >= E) return;

    const int   s  = src[e];
    const int   d  = dst[e];
    const float w  = ew[e];
    const int   c0 = lane * 4;

    float ea0 = bb[c0 + 0], ea1 = bb[c0 + 1], ea2 = bb[c0 + 2], ea3 = bb[c0 + 3];
#pragma unroll
    for (int k = 0; k < 11; ++k) {
        const float av = eattr[e * 11 + k];   // broadcast across the wave
        ea0 = fmaf(av, Wb[k * EMB + c0 + 0], ea0);
        ea1 = fmaf(av, Wb[k * EMB + c0 + 1], ea1);
        ea2 = fmaf(av, Wb[k * EMB + c0 + 2], ea2);
        ea3 = fmaf(av, Wb[k * EMB + c0 + 3], ea3);
    }
    const float4 hv = *(const float4*)(H + s * EMB + c0);
    float* p = AGG + d * EMB + c0;
    atomicAdd(p + 0, (hv.x + ea0) * w);
    atomicAdd(p + 1, (hv.y + ea1) * w);
    atomicAdd(p + 2, (hv.z + ea2) * w);
    atomicAdd(p + 3, (hv.w + ea3) * w);
}

// ------------------------------- BatchNorm stats: sum & sumsq per channel
__global__ void __launch_bounds__(128)
bn_stats(const float* __restrict__ H, float* __restrict__ stats, int M)
{
    const int c = threadIdx.x;            // 128 channels
    float s = 0.0f, s2 = 0.0f;
    for (int r = blockIdx.x; r < M; r += gridDim.x) {
        const float v = H[r * EMB + c];
        s += v;
        s2 = fmaf(v, v, s2);
    }
    atomicAdd(&stats[c], s);
    atomicAdd(&stats[EMB + c], s2);
}

// --------- BN apply + optional ReLU + fused global_add_pool into xpool
__global__ void __launch_bounds__(256)
bn_apply(const float* __restrict__ Hin, float* __restrict__ Hout,
         const float* __restrict__ stats,
         const float* __restrict__ gamma, const float* __restrict__ beta,
         const int* __restrict__ batch, float* __restrict__ xpool,
         int layer, int M, int do_relu)
{
    const int idx = blockIdx.x * blockDim.x + threadIdx.x;
    if (idx >= M * EMB) return;
    const int r = idx >> 7;
    const int c = idx & (EMB - 1);

    const float inv_n = 1.0f / (float)M;
    const float mu  = stats[c] * inv_n;
    const float var = stats[EMB + c] * inv_n - mu * mu;   // biased variance
    float y = gamma[c] * (Hin[idx] - mu) * rsqrtf(var + BN_EPS) + beta[c];
    if (do_relu) y = fmaxf(y, 0.0f);
    Hout[idx] = y;
    atomicAdd(&xpool[batch[r] * (LAYERS * EMB) + layer * EMB + c], y);
}

// ---------------------------------------------------------------- launcher
extern "C" void kernel_launch(void* const* d_in, const int* in_sizes, int n_in,
                              void* d_out, int out_size, void* d_ws, size_t ws_size,
                              hipStream_t stream) {
    const int*   batch  = (const int*)  d_in[0];
    const float* x      = (const float*)d_in[1];
    const int*   eidx   = (const int*)  d_in[2];   // [2,E]: row0=src, row1=dst
    const float* eattr  = (const float*)d_in[3];
    const float* ew     = (const float*)d_in[4];
    const float* W_atom = (const float*)d_in[5];
    const float* b_atom = (const float*)d_in[6];
    const float* W_bond = (const float*)d_in[7];
    const float* b_bond = (const float*)d_in[8];
    const float* Wl     = (const float*)d_in[9];   // [5,128,128]
    const float* bl     = (const float*)d_in[10];  // [5,128]
    const float* Wr     = (const float*)d_in[11];  // [5,128,128]
    const float* gamma  = (const float*)d_in[12];  // [5,128]
    const float* beta   = (const float*)d_in[13];  // [5,128]

    const int* src = eidx;
    const int* dst = eidx + N_EDGES;

    // workspace layout
    char* base = (char*)d_ws;
    const size_t hbytes = (size_t)N_NODES * EMB * sizeof(float);   // 25.6 MB
    float* hA    = (float*)(base);
    float* hB    = (float*)(base + hbytes);
    float* agg   = (float*)(base + 2 * hbytes);
    float* invd  = (float*)(base + 3 * hbytes);                    // 50000 f32
    float* stats = (float*)(base + 3 * hbytes + 262144);           // 256 f32

    float* xpool = (float*)d_out;                                  // [128, 640]
    float* h_out = (float*)d_out + N_GRAPHS * LAYERS * EMB;        // [N, 128]

    const int NE   = N_NODES * EMB;            // 6,400,000
    const int MT   = N_NODES / 16;             // 3125 tiles

    // 0) zero xpool and degree buffer
    zero_f32<<<(N_GRAPHS * LAYERS * EMB + 255) / 256, 256, 0, stream>>>(
        xpool, N_GRAPHS * LAYERS * EMB);
    zero_f32<<<(N_NODES + 255) / 256, 256, 0, stream>>>(invd, N_NODES);

    // 1) degrees -> reciprocal of mean denominator
    deg_kernel<<<(N_EDGES + 255) / 256, 256, 0, stream>>>(dst, invd, N_EDGES);
    invdenom_kernel<<<(N_NODES + 255) / 256, 256, 0, stream>>>(invd, N_NODES);

    // 2) node encoder (WMMA fp32)
    node_encoder_gemm<<<MT, 256, 0, stream>>>(x, W_atom, b_atom, hA);

    // 3) layers
    float* hin = hA;
    float* hot = hB;
    for (int i = 0; i < LAYERS; ++i) {
        zero_f32<<<(NE + 255) / 256, 256, 0, stream>>>(agg, NE);
        scatter_msgs<<<(N_EDGES * 32 + 255) / 256, 256, 0, stream>>>(
            src, dst, eattr, ew, W_bond, b_bond, hin, agg, N_EDGES);
        layer_gemm<<<MT, 256, 0, stream>>>(
            agg, invd, hin,
            Wl + (size_t)i * EMB * EMB, Wr + (size_t)i * EMB * EMB,
            bl + i * EMB, hot);
        zero_f32<<<1, 256, 0, stream>>>(stats, 2 * EMB);
        bn_stats<<<512, 128, 0, stream>>>(hot, stats, N_NODES);

        float* bn_dst = (i == LAYERS - 1) ? h_out : hot;   // last layer -> d_out
        bn_apply<<<(NE + 255) / 256, 256, 0, stream>>>(
            hot, bn_dst, stats, gamma + i * EMB, beta + i * EMB,
            batch, xpool, i, N_NODES, (i < LAYERS - 1) ? 1 : 0);

        // swap ping-pong buffers (post-BN activations live in bn_dst)
        float* t = hin; hin = bn_dst; hot = (t == hA || t == hB) ? t : hot;
        if (i < LAYERS - 1) { hot = (hin == hA) ? hB : hA; }
    }
}